// PigDetector_74766790689134
// MI455X (gfx1250) — compile-verified
//
#include <hip/hip_runtime.h>

// Problem constants (match reference)
#define B_   32
#define G_   2704
#define A_   3
#define C_   6
#define D_   1024
#define HID_ 512
#define L_   20
#define K_   128

typedef __attribute__((ext_vector_type(2))) float v2f;
typedef __attribute__((ext_vector_type(8))) float v8f;

// LDS layout strides (floats)
#define AS_STRIDE 34          // 128 rows x 32 k, padded to even -> 8B-aligned b64 frags
#define BS_STRIDE 160         // 16 k-pairs x 64 n (float2), pad = 32 banks -> conflict-free

// ---------------------------------------------------------------------------
// Kernel 1: objectness score + exact top-K selection, ascending index order.
// One block per batch; scores + flags live entirely in LDS.
// rank(g) = #{g' : s[g']>s[g]} + #{g'<g : s[g']==s[g]}  (strict total order)
// ---------------------------------------------------------------------------
__global__ void topk_kernel(const float* __restrict__ boxes, int* __restrict__ idx)
{
    __shared__ float         s_score[G_];
    __shared__ unsigned char s_flag[G_];
    const int b   = blockIdx.x;
    const int tid = threadIdx.x;
    const float* bb = boxes + (size_t)b * G_ * (A_ * C_);

    for (int g = tid; g < G_; g += blockDim.x) {
        const float* p = bb + (size_t)g * (A_ * C_);
        // mean over anchors of channel 4: elements at a*6+4 = {4,10,16}
        s_score[g] = (p[4] + p[10] + p[16]) * (1.0f / 3.0f);
    }
    __syncthreads();

    for (int g = tid; g < G_; g += blockDim.x) {
        const float v = s_score[g];
        int rank = 0;
        for (int g2 = 0; g2 < G_; ++g2) {
            const float u = s_score[g2];
            rank += (u > v) || (u == v && g2 < g);
        }
        s_flag[g] = (rank < K_) ? 1 : 0;
    }
    __syncthreads();

    // Compact selected indices in ascending grid order (matches jnp.sort(idx)).
    for (int g = tid; g < G_; g += blockDim.x) {
        if (s_flag[g]) {
            int pos = 0;
            for (int g2 = 0; g2 < g; ++g2) pos += s_flag[g2];
            idx[b * K_ + pos] = g;
        }
    }
}

// ---------------------------------------------------------------------------
// Kernel 2: gather selected boxes -> d_out tail region [B,K,A,C]
// ---------------------------------------------------------------------------
__global__ void gather_boxes_kernel(const float* __restrict__ boxes,
                                    const int* __restrict__ idx,
                                    float* __restrict__ out2)
{
    const int e = blockIdx.x * blockDim.x + threadIdx.x;
    const int total = B_ * K_ * A_ * C_;
    if (e >= total) return;
    const int r = e % (A_ * C_);
    const int m = (e / (A_ * C_)) % K_;
    const int b = e / (K_ * A_ * C_);
    const int g = idx[b * K_ + m];
    out2[e] = boxes[((size_t)b * G_ + g) * (A_ * C_) + r];
}

// ---------------------------------------------------------------------------
// Kernel 3: WMMA fp32 GEMM  C[b, row_off+m, n] = A[b,m,:] @ Bw[:,n] + bias[n]
//   GATHER=true : A[m][k] = feature[b, k, idx[b*K+m]]   (fused gather)
//   GATHER=false: A[m][k] = Abase[b*strideA + m*Kdim + k]
// Block: 256 threads (8 wave32), computes 128x64 of C; K staged in LDS by 32.
// A frags: single ds_load_b64 (even offsets, 136B row stride).
// B frags: pair-interleaved float2 LDS tiles -> single ds_load_b64, no movs.
// ---------------------------------------------------------------------------
template <bool GATHER>
__global__ void gemm_wmma_kernel(const float* __restrict__ Abase,
                                 const int*   __restrict__ idx,
                                 const float* __restrict__ Bw,
                                 const float* __restrict__ bias,
                                 float*       __restrict__ out,
                                 int M_real, int Kdim, long strideA, int row_off)
{
    __shared__ __align__(16) float As2[128 * AS_STRIDE];  // A[m][kk] at m*34+kk
    __shared__ __align__(16) float Bs2[16 * BS_STRIDE];   // pair p,n at p*160+2n
    __shared__ int s_idx[K_];

    const int b      = blockIdx.z;
    const int n0     = blockIdx.x * 64;
    const int tid    = threadIdx.x;
    const int lane   = tid & 31;
    const int wave   = tid >> 5;
    const int r      = lane & 15;     // row/col within fragment
    const int hi     = lane >> 4;     // half-wave: K pair {0,1} vs {2,3}
    const int m_base = wave * 16;

    const float* Ab = Abase + (size_t)b * (size_t)strideA;

    if (GATHER) {
        for (int i = tid; i < K_; i += 256) s_idx[i] = idx[b * K_ + i];
    }
    v8f acc0 = {}, acc1 = {}, acc2 = {}, acc3 = {};
    __syncthreads();

    for (int k0 = 0; k0 < Kdim; k0 += 32) {
        // ---- stage A tile: 128 rows x 32 k (4096 elems, 16 per thread) ----
#pragma unroll
        for (int j = 0; j < 16; ++j) {
            const int lin = tid + j * 256;
            const int m   = lin >> 5;
            const int kk  = lin & 31;
            float v = 0.0f;
            if (m < M_real) {
                if (GATHER) v = Ab[(size_t)(k0 + kk) * G_ + s_idx[m]];
                else        v = Ab[(size_t)m * Kdim + (k0 + kk)];
            }
            As2[m * AS_STRIDE + kk] = v;
        }
        // ---- stage B tile: 16 k-pairs x 64 n, pair-interleaved (4/thread) ----
#pragma unroll
        for (int j = 0; j < 4; ++j) {
            const int lin = tid + j * 256;      // 0..1023 pairs
            const int p   = lin >> 6;           // k-pair index 0..15
            const int nn  = lin & 63;
            v2f pr;
            pr.x = Bw[(size_t)(k0 + 2 * p + 0) * HID_ + (n0 + nn)];
            pr.y = Bw[(size_t)(k0 + 2 * p + 1) * HID_ + (n0 + nn)];
            *reinterpret_cast<v2f*>(&Bs2[p * BS_STRIDE + 2 * nn]) = pr;
        }
        if (k0 + 32 < Kdim)  // hint next weight chunk toward the caches
            __builtin_prefetch(&Bw[(size_t)(k0 + 32) * HID_ + n0], 0, 0);
        __syncthreads();

        // ---- compute: 8 k-steps of 4, 4 N-subtiles per wave ----
#pragma unroll
        for (int kk = 0; kk < 32; kk += 4) {
            // A 16x4 frag: VGPR0 = K{0|2}, VGPR1 = K{1|3}; one b64 load
            const v2f a = *reinterpret_cast<const v2f*>(
                &As2[(m_base + r) * AS_STRIDE + kk + 2 * hi]);
            const int p = (kk >> 1) + hi;       // k-pair row for this half-wave
            const v2f b0 = *reinterpret_cast<const v2f*>(&Bs2[p * BS_STRIDE + 2 * ( 0 + r)]);
            const v2f b1 = *reinterpret_cast<const v2f*>(&Bs2[p * BS_STRIDE + 2 * (16 + r)]);
            const v2f b2 = *reinterpret_cast<const v2f*>(&Bs2[p * BS_STRIDE + 2 * (32 + r)]);
            const v2f b3 = *reinterpret_cast<const v2f*>(&Bs2[p * BS_STRIDE + 2 * (48 + r)]);
            acc0 = __builtin_amdgcn_wmma_f32_16x16x4_f32(false, a, false, b0, (short)0, acc0, false, false);
            acc1 = __builtin_amdgcn_wmma_f32_16x16x4_f32(false, a, false, b1, (short)0, acc1, false, false);
            acc2 = __builtin_amdgcn_wmma_f32_16x16x4_f32(false, a, false, b2, (short)0, acc2, false, false);
            acc3 = __builtin_amdgcn_wmma_f32_16x16x4_f32(false, a, false, b3, (short)0, acc3, false, false);
        }
        __syncthreads();
    }

    // ---- epilogue: C layout VGPR i -> rows i (lanes 0-15) / i+8 (16-31) ----
#define EPILOGUE(ACC, NT)                                                      \
    _Pragma("unroll")                                                          \
    for (int i = 0; i < 8; ++i) {                                              \
        const int m = m_base + i + 8 * hi;                                     \
        if (m < M_real) {                                                      \
            const int n = n0 + (NT) * 16 + r;                                  \
            out[((size_t)b * (K_ + L_) + row_off + m) * HID_ + n] =            \
                (ACC)[i] + bias[n];                                            \
        }                                                                      \
    }
    EPILOGUE(acc0, 0)
    EPILOGUE(acc1, 1)
    EPILOGUE(acc2, 2)
    EPILOGUE(acc3, 3)
#undef EPILOGUE
}

// ---------------------------------------------------------------------------
extern "C" void kernel_launch(void* const* d_in, const int* in_sizes, int n_in,
                              void* d_out, int out_size, void* d_ws, size_t ws_size,
                              hipStream_t stream)
{
    const float* boxes   = (const float*)d_in[0];
    const float* feature = (const float*)d_in[1];
    const float* text    = (const float*)d_in[2];
    const float* W_vs    = (const float*)d_in[3];
    const float* b_vs    = (const float*)d_in[4];
    const float* W_tx    = (const float*)d_in[5];
    const float* b_tx    = (const float*)d_in[6];
    // d_in[7] = select_num (==128, compiled in as K_)
    (void)in_sizes; (void)n_in; (void)out_size; (void)ws_size;

    float* out  = (float*)d_out;
    float* out2 = out + (size_t)B_ * (K_ + L_) * HID_;  // sel_boxes region
    int*   idx  = (int*)d_ws;                           // B*K ints (16 KB)

    // 1) scores + top-K (ascending) per batch
    topk_kernel<<<B_, 256, 0, stream>>>(boxes, idx);

    // 2) gather selected boxes
    gather_boxes_kernel<<<(B_ * K_ * A_ * C_ + 255) / 256, 256, 0, stream>>>(
        boxes, idx, out2);

    // 3) linear_feature: fused-gather WMMA GEMM [128x1024]x[1024x512] per batch
    gemm_wmma_kernel<true><<<dim3(HID_ / 64, 1, B_), 256, 0, stream>>>(
        feature, idx, W_vs, b_vs, out, K_, D_, (long)D_ * G_, 0);

    // 4) linear_text: WMMA GEMM [20x512]x[512x512] per batch (rows 128..147)
    gemm_wmma_kernel<false><<<dim3(HID_ / 64, 1, B_), 256, 0, stream>>>(
        text, nullptr, W_tx, b_tx, out, L_, HID_, (long)L_ * HID_, K_);
}